// NEATNetwork_46746424050090
// MI455X (gfx1250) — compile-verified
//
#include <hip/hip_runtime.h>
#include <stdint.h>

// ---- problem constants (from reference) ----
#define N_IN     512
#define HID      1024
#define NLAYER   9          // 8 hidden + 1 output layer
#define N_OUT    256
#define DEG      64
#define BATCH    1024
#define W_TOT    (N_IN + 8 * HID + N_OUT)   // 8960 total activation columns
// ---- tiling ----
#define ROWSTRIDE 8968      // 8960 + 8 pad dwords; 8968 % 64 == 8 -> rows land in different banks
#define TB        8         // batch rows per workgroup: 8 * 8968 * 4B = 286,976 B LDS (<= 320KB WGP LDS)
#define NTHREADS  512       // 16 wave32 waves

#define AS1 __attribute__((address_space(1)))
#define AS3 __attribute__((address_space(3)))

// 16-byte int vector type matching the builtin's expected pointee
typedef int v4i_t __attribute__((__vector_size__(16)));

// ---- CDNA5 async global<->LDS DMA (ASYNCcnt-tracked), with safe fallbacks ----
__device__ __forceinline__ void async_g2l_b128(const void* g, void* l) {
#if __has_builtin(__builtin_amdgcn_global_load_async_to_lds_b128)
    __builtin_amdgcn_global_load_async_to_lds_b128(
        (AS1 v4i_t*)(uintptr_t)g,
        (AS3 v4i_t*)(uint32_t)(uintptr_t)l,  // low 32 bits of generic LDS ptr == as(3) offset
        0, 0);
#else
    *(float4*)l = *(const float4*)g;
#endif
}

__device__ __forceinline__ void async_l2g_b128(void* g, const void* l) {
#if __has_builtin(__builtin_amdgcn_global_store_async_from_lds_b128)
    __builtin_amdgcn_global_store_async_from_lds_b128(
        (AS1 v4i_t*)(uintptr_t)g,
        (AS3 v4i_t*)(uint32_t)(uintptr_t)l,
        0, 0);
#else
    *(float4*)g = *(const float4*)l;
#endif
}

__device__ __forceinline__ void wait_async_all() {
#if __has_builtin(__builtin_amdgcn_s_wait_asynccnt)
    __builtin_amdgcn_s_wait_asynccnt(0);
#else
    asm volatile("s_wait_asynccnt 0x0" ::: "memory");
#endif
}

__global__ __launch_bounds__(NTHREADS)
void neat_lds_resident_kernel(const float* __restrict__ in,
                              const float* __restrict__ wts,
                              const int*   __restrict__ sidx,
                              float*       __restrict__ out) {
    // Full activation history for TB batch rows lives in LDS (CDNA5: 320KB/WG).
    __shared__ float acts[TB * ROWSTRIDE];

    const int tid  = threadIdx.x;
    const int row0 = blockIdx.x * TB;

    // ---- stage inputs (TB x N_IN f32 = 16KB) into LDS via async DMA ----
    {
        const int xfers = TB * N_IN / 4;                 // 1024 float4 transfers
        for (int t = tid; t < xfers; t += NTHREADS) {
            const int r  = t >> 7;                       // N_IN/4 = 128 xfers per row
            const int c4 = t & 127;
            async_g2l_b128(in + (size_t)(row0 + r) * N_IN + c4 * 4,
                           &acts[r * ROWSTRIDE + c4 * 4]);
        }
        wait_async_all();
    }
    __syncthreads();

    // wave32 lane layout: lane = (row r in 0..7, node-sub q in 0..3)
    const int lane = tid & 31;
    const int wid  = tid >> 5;                            // 16 waves
    const int r    = lane & 7;
    const int q    = lane >> 3;
    const float* __restrict__ arow = &acts[r * ROWSTRIDE];

    size_t eoff = 0;       // edge offset into weights / src_idx
    int    col0 = N_IN;    // column where this layer's outputs are appended

    #pragma unroll 1
    for (int layer = 0; layer < NLAYER; ++layer) {
        const int n = (layer < NLAYER - 1) ? HID : N_OUT;

        // each wave handles 4 nodes x 8 rows per step; 16 waves -> 64 nodes/step
        for (int j0 = wid * 4; j0 < n; j0 += (NTHREADS / 32) * 4) {
            const int j = j0 + q;
            const int4*   ip = (const int4*)  (sidx + eoff + (size_t)j * DEG);
            const float4* wp = (const float4*)(wts  + eoff + (size_t)j * DEG);

            // speculative prefetch of next step's idx/weights (64 nodes ahead = +16KB)
            __builtin_prefetch((const char*)ip + 64 * DEG * 4, 0, 0);
            __builtin_prefetch((const char*)wp + 64 * DEG * 4, 0, 0);

            float acc = 0.0f;
            #pragma unroll 4
            for (int c = 0; c < DEG / 4; ++c) {
                const int4   i4 = ip[c];
                const float4 w4 = wp[c];
                acc += arow[i4.x] * w4.x;
                acc += arow[i4.y] * w4.y;
                acc += arow[i4.z] * w4.z;
                acc += arow[i4.w] * w4.w;
            }
            // sigmoid; lane writes its (row, node) result straight to LDS
            acts[r * ROWSTRIDE + col0 + j] = 1.0f / (1.0f + __expf(-acc));
        }
        __syncthreads();
        eoff += (size_t)n * DEG;
        col0 += n;
    }

    // ---- write last N_OUT columns (the final layer's outputs) via async LDS->global ----
    {
        const int xfers = TB * N_OUT / 4;                 // 512 float4 transfers
        const int ocol  = W_TOT - N_OUT;                  // 8704
        for (int t = tid; t < xfers; t += NTHREADS) {
            const int rr = t >> 6;                        // N_OUT/4 = 64 xfers per row
            const int c4 = t & 63;
            async_l2g_b128(out + (size_t)(row0 + rr) * N_OUT + c4 * 4,
                           &acts[rr * ROWSTRIDE + ocol + c4 * 4]);
        }
        wait_async_all();                                 // drain ASYNCcnt before endpgm
    }
}

extern "C" void kernel_launch(void* const* d_in, const int* in_sizes, int n_in,
                              void* d_out, int out_size, void* d_ws, size_t ws_size,
                              hipStream_t stream) {
    const float* in   = (const float*)d_in[0];   // (1024, 512) f32
    const float* wts  = (const float*)d_in[1];   // (540672,)   f32
    const int*   sidx = (const int*)  d_in[2];   // (540672,)   i32
    float*       out  = (float*)d_out;           // (1024, 256) f32

    dim3 grid(BATCH / TB);    // 128 workgroups (~1 per WGP, LDS-limited anyway)
    dim3 block(NTHREADS);
    hipLaunchKernelGGL(neat_lds_resident_kernel, grid, block, 0, stream,
                       in, wts, sidx, out);
}